// VAELoss_50921132262091
// MI455X (gfx1250) — compile-verified
//
#include <hip/hip_runtime.h>
#include <hip/hip_bf16.h>
#include <math.h>

// Problem constants
#define BB 2048
#define SS 277
#define RR 80
#define LL 12
#define ZZ 56

typedef __attribute__((ext_vector_type(2))) float v2f;
typedef __attribute__((ext_vector_type(8))) float v8f;

__device__ __forceinline__ float wave_sum(float v) {
#pragma unroll
  for (int off = 16; off >= 1; off >>= 1) v += __shfl_xor(v, off, 32);
  return v;
}

__device__ __forceinline__ int wave_max(int v) {
#pragma unroll
  for (int off = 16; off >= 1; off >>= 1) {
    int o = __shfl_xor(v, off, 32);
    v = (o > v) ? o : v;
  }
  return v;
}

// ---------------------------------------------------------------------------
// Kernel 1: BCE over all B*S rows. One wave32 per row, grid-strided.
// Row = 80 elems: lanes cover [lane], [lane+32]; the 16-wide tail chunk is
// loaded by ALL lanes (lanes>=16 duplicate the same cacheline — free) and
// neutralized via m2=0 so there are no divergent global loads in the loop.
// HBM-bound: streams x (181.5MB) + recon_x (181.5MB); masks/lhs in LDS.
// ---------------------------------------------------------------------------
__global__ __launch_bounds__(256) void bce_kernel(const float* __restrict__ x,
                                                  const float* __restrict__ recon,
                                                  const float* __restrict__ masks,
                                                  const int* __restrict__ lhs,
                                                  float* __restrict__ partials) {
  __shared__ float smasks[LL * RR];
  __shared__ int slhs[RR];
  __shared__ float red[256];
  for (int i = threadIdx.x; i < LL * RR; i += blockDim.x) smasks[i] = masks[i];
  for (int i = threadIdx.x; i < RR; i += blockDim.x) slhs[i] = lhs[i];
  __syncthreads();

  const int lane = threadIdx.x & 31;
  const int laneM = lane & 15;
  const int wid = (int)((blockIdx.x * blockDim.x + threadIdx.x) >> 5);
  const int nwaves = (int)((gridDim.x * blockDim.x) >> 5);
  const int rows = BB * SS;

  float acc = 0.f;
  for (int row = wid; row < rows; row += nwaves) {
    const size_t base = (size_t)row * RR;
    // Unconditional coalesced loads (tail chunk duplicated across half-waves)
    float xv0 = x[base + lane];
    float xv1 = x[base + 32 + lane];
    float xv2 = x[base + 64 + laneM];
    float rv0 = recon[base + lane];
    float rv1 = recon[base + 32 + lane];
    float rv2 = recon[base + 64 + laneM];

    // one-hot argmax (tail only counted by lanes < 16)
    int cand = -1;
    if (xv0 > 0.5f) cand = lane;
    if (xv1 > 0.5f) cand = lane + 32;
    if (lane < 16 && xv2 > 0.5f) cand = lane + 64;
    cand = wave_max(cand);  // broadcast to all lanes

    const float* m = &smasks[slhs[cand] * RR];
    float m0 = m[lane];
    float m1 = m[lane + 32];
    float m2 = (lane < 16) ? m[64 + laneM] : 0.f;  // kills duplicated tail

    float p0 = rv0 * m0, p1 = rv1 * m1, p2 = rv2 * m2;
    float inv = 1.f / wave_sum(p0 + p1 + p2);
    float o0 = p0 * inv, o1 = p1 * inv, o2 = p2 * inv;

    // x is exactly one-hot: only the true-rule element needs log(); the rest
    // need log1p(). Clamp at -100 before weighting (matches torch BCE clamp).
    // Lanes>=16: o2==0 and cand!=lane+64 (cand<=79), so t2 = log1p(0) = 0.
    float t0 = (cand == lane) ? fmaxf(__logf(o0), -100.f)
                              : fmaxf(log1pf(-o0), -100.f);
    float t1 = (cand == lane + 32) ? fmaxf(__logf(o1), -100.f)
                                   : fmaxf(log1pf(-o1), -100.f);
    float t2 = (cand == lane + 64) ? fmaxf(__logf(o2), -100.f)
                                   : fmaxf(log1pf(-o2), -100.f);
    acc += t0 + t1 + t2;
  }

  red[threadIdx.x] = acc;
  __syncthreads();
#pragma unroll
  for (int s = 128; s > 0; s >>= 1) {
    if ((int)threadIdx.x < s) red[threadIdx.x] += red[threadIdx.x + s];
    __syncthreads();
  }
  if (threadIdx.x == 0) partials[blockIdx.x] = red[0];
}

// ---------------------------------------------------------------------------
// Kernel 2: var = mu^T mu (56x2048x56 f32 GEMM) via V_WMMA_F32_16X16X4_F32,
// padded to 64x64. 16 waves, one 16x16 output tile each; K-loop step 4.
// A[M][K] = mu[K][M], B[K][N] = mu[K][N] -> identical per-lane load formula.
// Columns >= 56 are CLAMPED to 55 (valid finite loads, no exec branching);
// the garbage products land only in output rows/cols >= 56, which the
// abs-sum already excludes.
// ---------------------------------------------------------------------------
__global__ __launch_bounds__(512) void var_wmma_kernel(const float* __restrict__ mu,
                                                       float* __restrict__ partials) {
  const int lane = threadIdx.x & 31;
  const int wave = threadIdx.x >> 5;  // 0..15
  const int ti = wave >> 2;           // tile row
  const int tj = wave & 3;            // tile col
  const int half = lane >> 4;         // 0: K=k0,k0+1 ; 1: K=k0+2,k0+3
  const int laneM = lane & 15;
  const int colA = ti * 16 + laneM;   // var row index == mu column
  const int colB = tj * 16 + laneM;   // var col index == mu column
  const int colAc = (colA < ZZ) ? colA : (ZZ - 1);  // clamp: always valid
  const int colBc = (colB < ZZ) ? colB : (ZZ - 1);

  v8f c = {};
#pragma unroll 4
  for (int k0 = 0; k0 < BB; k0 += 4) {
    const int kA = k0 + 2 * half;
    v2f a, b;
    a.x = mu[(size_t)kA * ZZ + colAc];
    a.y = mu[(size_t)(kA + 1) * ZZ + colAc];
    b.x = mu[(size_t)kA * ZZ + colBc];
    b.y = mu[(size_t)(kA + 1) * ZZ + colBc];
    c = __builtin_amdgcn_wmma_f32_16x16x4_f32(
        /*neg_a=*/false, a, /*neg_b=*/false, b,
        /*c_mod=*/(short)0, c, /*reuse_a=*/false, /*reuse_b=*/false);
  }

  // C/D layout: VGPR v -> M = v (lanes 0-15) or v+8 (lanes 16-31); N = lane%16
  float s = 0.f;
#pragma unroll
  for (int v = 0; v < 8; ++v) {
    int gr = ti * 16 + v + (half ? 8 : 0);
    int gc = colB;
    if (gr < ZZ && gc < ZZ) {
      float val = c[v] * (1.0f / BB) - ((gr == gc) ? 1.0f : 0.0f);
      s += fabsf(val);
    }
  }
  s = wave_sum(s);
  if (lane == 0) partials[wave] = s;
}

// ---------------------------------------------------------------------------
// Kernel 3: reduce partials, compute avg_mu moment, emit scalar loss.
// ---------------------------------------------------------------------------
__global__ __launch_bounds__(256) void final_kernel(const float* __restrict__ mu,
                                                    const float* __restrict__ bce_partials,
                                                    const float* __restrict__ var_partials,
                                                    float* __restrict__ out) {
  __shared__ float red[256];
  const int tid = threadIdx.x;

  // BCE partials (1024)
  float b = 0.f;
  for (int i = tid; i < 1024; i += 256) b += bce_partials[i];
  red[tid] = b;
  __syncthreads();
#pragma unroll
  for (int s = 128; s > 0; s >>= 1) {
    if (tid < s) red[tid] += red[tid + s];
    __syncthreads();
  }
  float bce_sum = red[0];
  __syncthreads();

  // |var - I| partials (16)
  red[tid] = (tid < 16) ? var_partials[tid] : 0.f;
  __syncthreads();
#pragma unroll
  for (int s = 128; s > 0; s >>= 1) {
    if (tid < s) red[tid] += red[tid + s];
    __syncthreads();
  }
  float abs_sum = red[0];
  __syncthreads();

  // avg_mu^2 : 56 column means over 2048 rows (L2-resident)
  float a2 = 0.f;
  if (tid < ZZ) {
    float cs = 0.f;
    for (int k = 0; k < BB; ++k) cs += mu[(size_t)k * ZZ + tid];
    float a = cs / (float)BB;
    a2 = a * a;
  }
  red[tid] = a2;
  __syncthreads();
#pragma unroll
  for (int s = 128; s > 0; s >>= 1) {
    if (tid < s) red[tid] += red[tid + s];
    __syncthreads();
  }
  if (tid == 0) {
    float bce = -bce_sum / (float)(SS * BB);
    float mom = red[0] / (float)ZZ + abs_sum / (float)(ZZ * ZZ);
    out[0] = bce + mom;
  }
}

extern "C" void kernel_launch(void* const* d_in, const int* in_sizes, int n_in,
                              void* d_out, int out_size, void* d_ws, size_t ws_size,
                              hipStream_t stream) {
  const float* x = (const float*)d_in[0];       // [B,S,R] one-hot f32
  const float* recon = (const float*)d_in[1];   // [B,S,R] f32
  const float* mu = (const float*)d_in[2];      // [B,Z] f32
  // d_in[3] = log_var: unused by the reference expression
  const float* masks = (const float*)d_in[4];   // [L,R] f32
  const int* lhs = (const int*)d_in[5];         // [R] i32
  float* out = (float*)d_out;
  float* ws = (float*)d_ws;                     // [0..1023] bce, [1024..1039] var

  bce_kernel<<<1024, 256, 0, stream>>>(x, recon, masks, lhs, ws);
  var_wmma_kernel<<<1, 512, 0, stream>>>(mu, ws + 1024);
  final_kernel<<<1, 256, 0, stream>>>(mu, ws, ws + 1024, out);
}